// CausalGP_2087354105999
// MI455X (gfx1250) — compile-verified
//
#include <hip/hip_runtime.h>
#include <hip/hip_bf16.h>

typedef __attribute__((ext_vector_type(2))) float v2f;
typedef __attribute__((ext_vector_type(8))) float v8f;

#define B_   8
#define M_   1024
#define P_   256
#define N_   4096
#define D_   257
#define ROWS (B_ * M_)   // 8192 query points
#define TM   256         // block tile rows
#define TN   128         // block tile cols
#define KT   16          // K per LDS stage
#define LDA  20          // LDS stride for A tile rows (16B aligned, conflict-free fan-out)
#define LDB  128         // B tile cols

// fp32 WMMA: D(16x16) = A(16x4) * B(4x16) + C
__device__ __forceinline__ v8f wmma4(v2f a, v2f b, v8f c) {
  return __builtin_amdgcn_wmma_f32_16x16x4_f32(
      /*neg_a=*/false, a, /*neg_b=*/false, b,
      /*c_mod=*/(short)0, c, /*reuse_a=*/false, /*reuse_b=*/false);
}

__device__ __forceinline__ float rowsum16(float t) {
  // sum over the 16 lanes of this half-wave (masks < 16 never cross halves)
  t += __shfl_xor(t, 1, 32);
  t += __shfl_xor(t, 2, 32);
  t += __shfl_xor(t, 4, 32);
  t += __shfl_xor(t, 8, 32);
  return t;
}

// ---------------------------------------------------------------- init d_out
__global__ void init_out_k(float* __restrict__ out) {
  int i = blockIdx.x * 256 + threadIdx.x;
  if (i < ROWS) {
    out[i]        = 0.0f;  // f_loc accumulator
    out[ROWS + i] = 1.0f;  // f_var = Kxx diag (=1) minus quadratic form
  }
}

// ------------------------------------------------- row norms ||XA[b,m,:]||^2
__global__ void xnorm_k(const float* __restrict__ X, const int* __restrict__ A,
                        float* __restrict__ xn) {
  int row = blockIdx.x;                    // 0..8191
  float v = X[(size_t)row * P_ + threadIdx.x];
  float p = v * v;
  for (int m = 16; m >= 1; m >>= 1) p += __shfl_xor(p, m, 32);
  __shared__ float red[8];
  if ((threadIdx.x & 31) == 0) red[threadIdx.x >> 5] = p;
  __syncthreads();
  if (threadIdx.x == 0) {
    float s = 0.f;
    for (int i = 0; i < 8; ++i) s += red[i];
    float a = (float)A[row];
    xn[row] = s + a * a;                   // appended causal bit contributes a^2
  }
}

// ---------------------------------------------- row norms ||XA_train[n,:]||^2
__global__ void znorm_k(const float* __restrict__ Z, float* __restrict__ zn) {
  int row = blockIdx.x;                    // 0..4095
  float v = Z[(size_t)row * D_ + threadIdx.x];
  float p = v * v;
  if (threadIdx.x == 0) {                  // 257th element
    float t = Z[(size_t)row * D_ + 256];
    p += t * t;
  }
  for (int m = 16; m >= 1; m >>= 1) p += __shfl_xor(p, m, 32);
  __shared__ float red[8];
  if ((threadIdx.x & 31) == 0) red[threadIdx.x >> 5] = p;
  __syncthreads();
  if (threadIdx.x == 0) {
    float s = 0.f;
    for (int i = 0; i < 8; ++i) s += red[i];
    zn[row] = s;
  }
}

// ---------------------------------------------------------------------------
// GEMM1: Kxz[r,n] = exp(-0.5*max(xn[r]+zn[n]-2*<XA_r, Z_n>, 0)),
// fused f_loc[r] += sum_n Kxz[r,n]*alpha[n].
// 256x128 tile, double-buffered LDS; K=257 (256 via WMMA + rank-1 tail).
// ---------------------------------------------------------------------------
__global__ __launch_bounds__(256) void gemm1_rbf_k(
    const float* __restrict__ X, const int* __restrict__ A,
    const float* __restrict__ Z, const float* __restrict__ alpha,
    const float* __restrict__ xn, const float* __restrict__ zn,
    float* __restrict__ Kxz, float* __restrict__ out) {
  __shared__ float As[2][TM * LDA];       // row-major rows, stride 20
  __shared__ float Bs[2][8 * LDB * 2];    // K-pair interleaved: [kp][n] = {k,k+1}
  const int tid  = threadIdx.x;
  const int lane = tid & 31, wave = tid >> 5;
  const int l16  = lane & 15, h = lane >> 4;
  const int wr   = wave >> 1, wc = wave & 1;      // 4x2 wave grid -> 64x64 each
  const int gr0  = blockIdx.y * TM;
  const int gc0  = blockIdx.x * TN;

  v8f acc[4][4] = {};
  float4 aR[4];
  float2 bR[4];

  // staging maps (A: 1024 float4 slots; B: 1024 float2 slots)
  int arr[4], akk[4], bkp[4], bcc[4];
#pragma unroll
  for (int e = 0; e < 4; ++e) {
    int idx = e * 256 + tid;
    arr[e] = idx >> 2;           // 0..255 tile row
    akk[e] = (idx & 3) * 4;      // 0,4,8,12
    bkp[e] = idx >> 7;           // 0..7 K-pair
    bcc[e] = idx & 127;          // tile col
  }

  auto loadRegs = [&](int k0) {
#pragma unroll
    for (int e = 0; e < 4; ++e)
      aR[e] = *(const float4*)(X + (size_t)(gr0 + arr[e]) * P_ + k0 + akk[e]);
#pragma unroll
    for (int e = 0; e < 4; ++e) {
      const float* zp = Z + (size_t)(gc0 + bcc[e]) * D_ + k0 + 2 * bkp[e];
      bR[e] = make_float2(zp[0], zp[1]);      // gather-transpose
    }
  };
  auto storeLDS = [&](int buf) {
#pragma unroll
    for (int e = 0; e < 4; ++e)
      *(float4*)(&As[buf][arr[e] * LDA + akk[e]]) = aR[e];
#pragma unroll
    for (int e = 0; e < 4; ++e)
      *(float2*)(&Bs[buf][(bkp[e] * LDB + bcc[e]) * 2]) = bR[e];
  };
  auto compute = [&](int buf) {
#pragma unroll
    for (int ks = 0; ks < 4; ++ks) {
      const int kb = ks * 4 + 2 * h;            // even -> b64 aligned
      v2f af[4], bf[4];
#pragma unroll
      for (int i = 0; i < 4; ++i)
        af[i] = *(const v2f*)(&As[buf][(wr * 64 + i * 16 + l16) * LDA + kb]);
#pragma unroll
      for (int j = 0; j < 4; ++j)
        bf[j] = *(const v2f*)(&Bs[buf][((kb >> 1) * LDB + wc * 64 + j * 16 + l16) * 2]);
#pragma unroll
      for (int i = 0; i < 4; ++i)
#pragma unroll
        for (int j = 0; j < 4; ++j)
          acc[i][j] = wmma4(af[i], bf[j], acc[i][j]);
    }
  };

  const int nt = P_ / KT;                       // 16 stages (K=256)
  loadRegs(0);
  storeLDS(0);
  int buf = 0;
  for (int t = 0; t < nt; ++t) {
    __syncthreads();
    if (t + 1 < nt) loadRegs((t + 1) * KT);     // overlap HBM with WMMA
    compute(buf);
    if (t + 1 < nt) storeLDS(buf ^ 1);
    buf ^= 1;
  }

  // epilogue: rank-1 tail (d=256), RBF, store Kxz, reduce f_loc
#pragma unroll
  for (int i = 0; i < 4; ++i)
#pragma unroll
    for (int j = 0; j < 4; ++j) {
      int gcol   = gc0 + wc * 64 + j * 16 + l16;
      float zt   = Z[(size_t)gcol * D_ + 256];
      float znv  = zn[gcol];
      float al   = alpha[gcol];
      v8f   d    = acc[i][j];
#pragma unroll
      for (int r = 0; r < 8; ++r) {
        int grow   = gr0 + wr * 64 + i * 16 + r + 8 * h;
        float aval = (float)A[grow];
        float S    = d[r] + aval * zt;          // complete the D=257 dot
        float sq   = fmaxf(xn[grow] + znv - 2.0f * S, 0.0f);
        float kv   = __expf(-0.5f * sq);
        Kxz[(size_t)grow * N_ + gcol] = kv;
        float t = rowsum16(kv * al);
        if (l16 == r) atomicAdd(out + grow, t); // one lane per row
      }
    }
}

// ---------------------------------------------------------------------------
// GEMM2 (dominant, 275 GFLOP): W = Kxz @ K_inv, fused with
// f_var[r] -= sum_n W[r,n]*Kxz[r,n]  (W never hits memory).
// ---------------------------------------------------------------------------
__global__ __launch_bounds__(256) void gemm2_var_k(
    const float* __restrict__ Kxz, const float* __restrict__ Kinv,
    float* __restrict__ out) {
  __shared__ float As[2][TM * LDA];
  __shared__ float Bs[2][8 * LDB * 2];
  const int tid  = threadIdx.x;
  const int lane = tid & 31, wave = tid >> 5;
  const int l16  = lane & 15, h = lane >> 4;
  const int wr   = wave >> 1, wc = wave & 1;
  const int gr0  = blockIdx.y * TM;
  const int gc0  = blockIdx.x * TN;

  v8f acc[4][4] = {};
  float4 aR[4];
  float4 b0R, b1R;

  int arr[4], akk[4];
#pragma unroll
  for (int e = 0; e < 4; ++e) {
    int idx = e * 256 + tid;
    arr[e] = idx >> 2;
    akk[e] = (idx & 3) * 4;
  }
  const int bkp = tid >> 5;                 // 0..7 K-pair
  const int bn4 = (tid & 31) * 4;           // col quad

  auto loadRegs = [&](int k0) {
#pragma unroll
    for (int e = 0; e < 4; ++e)
      aR[e] = *(const float4*)(Kxz + (size_t)(gr0 + arr[e]) * N_ + k0 + akk[e]);
    b0R = *(const float4*)(Kinv + (size_t)(k0 + 2 * bkp)     * N_ + gc0 + bn4);
    b1R = *(const float4*)(Kinv + (size_t)(k0 + 2 * bkp + 1) * N_ + gc0 + bn4);
  };
  auto storeLDS = [&](int buf) {
#pragma unroll
    for (int e = 0; e < 4; ++e)
      *(float4*)(&As[buf][arr[e] * LDA + akk[e]]) = aR[e];
    const float* p0 = (const float*)&b0R;
    const float* p1 = (const float*)&b1R;
#pragma unroll
    for (int e = 0; e < 4; ++e)
      *(float2*)(&Bs[buf][(bkp * LDB + bn4 + e) * 2]) = make_float2(p0[e], p1[e]);
  };
  auto compute = [&](int buf) {
#pragma unroll
    for (int ks = 0; ks < 4; ++ks) {
      const int kb = ks * 4 + 2 * h;
      v2f af[4], bf[4];
#pragma unroll
      for (int i = 0; i < 4; ++i)
        af[i] = *(const v2f*)(&As[buf][(wr * 64 + i * 16 + l16) * LDA + kb]);
#pragma unroll
      for (int j = 0; j < 4; ++j)
        bf[j] = *(const v2f*)(&Bs[buf][((kb >> 1) * LDB + wc * 64 + j * 16 + l16) * 2]);
#pragma unroll
      for (int i = 0; i < 4; ++i)
#pragma unroll
        for (int j = 0; j < 4; ++j)
          acc[i][j] = wmma4(af[i], bf[j], acc[i][j]);
    }
  };

  const int nt = N_ / KT;                   // 256 stages
  loadRegs(0);
  storeLDS(0);
  int buf = 0;
  for (int t = 0; t < nt; ++t) {
    __syncthreads();
    if (t + 1 < nt) loadRegs((t + 1) * KT);
    compute(buf);
    if (t + 1 < nt) storeLDS(buf ^ 1);
    buf ^= 1;
  }

  // epilogue: f_var[row] -= sum over this tile's cols of W*Kxz
#pragma unroll
  for (int i = 0; i < 4; ++i)
#pragma unroll
    for (int j = 0; j < 4; ++j) {
      int gcol = gc0 + wc * 64 + j * 16 + l16;
      v8f  d   = acc[i][j];
#pragma unroll
      for (int r = 0; r < 8; ++r) {
        int grow = gr0 + wr * 64 + i * 16 + r + 8 * h;
        float kv = Kxz[(size_t)grow * N_ + gcol];
        float t  = rowsum16(d[r] * kv);
        if (l16 == r) atomicAdd(out + ROWS + grow, -t);
      }
    }
}

extern "C" void kernel_launch(void* const* d_in, const int* in_sizes, int n_in,
                              void* d_out, int out_size, void* d_ws, size_t ws_size,
                              hipStream_t stream) {
  const float* X     = (const float*)d_in[0];   // [8,1024,256]
  const int*   A     = (const int*)d_in[1];     // [8,1024]
  const float* Z     = (const float*)d_in[2];   // XA_train [4096,257]
  const float* alpha = (const float*)d_in[3];   // [4096]
  const float* Kinv  = (const float*)d_in[4];   // [4096,4096]
  float* out = (float*)d_out;                   // f_loc[8192] ++ f_var[8192]

  float* ws  = (float*)d_ws;
  float* Kxz = ws;                              // 8192*4096 floats = 128 MB
  float* xn  = ws + (size_t)ROWS * N_;          // 8192
  float* zn  = xn + ROWS;                       // 4096

  init_out_k<<<(ROWS + 255) / 256, 256, 0, stream>>>(out);
  xnorm_k<<<ROWS, 256, 0, stream>>>(X, A, xn);
  znorm_k<<<N_, 256, 0, stream>>>(Z, zn);

  dim3 grid(N_ / TN, ROWS / TM);                // 32 x 32 tiles
  gemm1_rbf_k<<<grid, 256, 0, stream>>>(X, A, Z, alpha, xn, zn, Kxz, out);
  gemm2_var_k<<<grid, 256, 0, stream>>>(Kxz, Kinv, out);
}